// YoloLoss_61899068670264
// MI455X (gfx1250) — compile-verified
//
#include <hip/hip_runtime.h>
#include <math.h>

// YOLO loss: preds/targets (512, 28, 28, 28) f32 -> scalar.
// Memory-bound: ~90 MB read => ~3.9 us floor at 23.3 TB/s.
// Pass 1: one cell per thread (grid-stride), 7x float4 loads per tensor per
//         cell, per-block LDS tree reduction -> 1024 partials in d_ws.
// Pass 2: single wave32; lane partial sums, then cross-lane reduction via
//         V_WMMA_F32_16X16X4_F32 with B = ones (gfx1250 matrix path, f32).

#define NBATCH   512
#define SDIM     28
#define DDIM     28
#define NCLS     20
#define NCELLS   (NBATCH * SDIM * SDIM)      // 401408
#define P1_BLOCKS  1024
#define P1_THREADS 256
#define LAMBDA_COORD 2.0f

typedef float v2f __attribute__((ext_vector_type(2)));
typedef float v8f __attribute__((ext_vector_type(8)));

__global__ __launch_bounds__(P1_THREADS)
void yolo_pass1(const float* __restrict__ preds,
                const float* __restrict__ targets,
                float* __restrict__ partials) {
    const int tid    = threadIdx.x;
    const int gid    = blockIdx.x * P1_THREADS + tid;
    const int stride = gridDim.x * P1_THREADS;

    float acc = 0.0f;

    for (int c = gid; c < NCELLS; c += stride) {
        const float4* p4 = reinterpret_cast<const float4*>(preds   + (size_t)c * DDIM);
        const float4* t4 = reinterpret_cast<const float4*>(targets + (size_t)c * DDIM);

        float p[DDIM], t[DDIM];
        #pragma unroll
        for (int i = 0; i < 7; ++i) {
            float4 a = p4[i];
            p[4*i+0] = a.x; p[4*i+1] = a.y; p[4*i+2] = a.z; p[4*i+3] = a.w;
            float4 b = t4[i];
            t[4*i+0] = b.x; t[4*i+1] = b.y; t[4*i+2] = b.z; t[4*i+3] = b.w;
        }

        // Coord loss: channels 0..3 (box 0) and 24..27 (box 1).
        float cl = 0.0f;
        #pragma unroll
        for (int k = 0; k < 4; ++k) {
            float d0 = p[k]      - t[k];      cl += d0 * d0;
            float d1 = p[24 + k] - t[24 + k]; cl += d1 * d1;
        }

        // Class channels 8..27: mask = sum(tgt)>0, label = first argmax(tgt).
        float tsum = 0.0f, tmax = -INFINITY;
        int lab = 0;
        #pragma unroll
        for (int k = 0; k < NCLS; ++k) {
            float tv = t[8 + k];
            tsum += tv;
            if (tv > tmax) { tmax = tv; lab = k; }
        }

        // Stable log-softmax CE; gather p_cls[lab] with a select chain so the
        // p[] array stays register-resident (no dynamic indexing -> no scratch).
        float pmax = -INFINITY, psel = 0.0f;
        #pragma unroll
        for (int k = 0; k < NCLS; ++k) {
            float pv = p[8 + k];
            pmax = fmaxf(pmax, pv);
            psel = (k == lab) ? pv : psel;
        }
        float se = 0.0f;
        #pragma unroll
        for (int k = 0; k < NCLS; ++k)
            se += __expf(p[8 + k] - pmax);
        float ce = (pmax + __logf(se)) - psel;

        acc += LAMBDA_COORD * cl + ((tsum > 0.0f) ? ce : 0.0f);
    }

    // Deterministic per-block tree reduction.
    __shared__ float sm[P1_THREADS];
    sm[tid] = acc;
    __syncthreads();
    #pragma unroll
    for (int off = P1_THREADS / 2; off > 0; off >>= 1) {
        if (tid < off) sm[tid] += sm[tid + off];
        __syncthreads();
    }
    if (tid == 0) partials[blockIdx.x] = sm[0];
}

// Pass 2: one wave32. Lane i serially sums partials[i::32] (uniform trip
// count -> EXEC stays all-ones for WMMA). Cross-lane sum via
// v_wmma_f32_16x16x4_f32 with B = ones:
//   A[m][0] = s(lane m), A[m][2] = s(lane m+16), A[m][1]=A[m][3]=0
//   => D[m][n] = s(m) + s(m+16) for all n.
// C/D layout: lane L<16 VGPR r = D[r][L]; lane L+16 VGPR r = D[8+r][L].
// So sum of lane0's 8 regs + sum of lane16's 8 regs = total over 32 lanes.
__global__ __launch_bounds__(32)
void yolo_reduce(const float* __restrict__ partials, int n,
                 float* __restrict__ out, float inv_batch) {
    const int lane = threadIdx.x;

    float s = 0.0f;
    for (int i = lane; i < n; i += 32)   // n multiple of 32 -> uniform
        s += partials[i];

    v2f a; a.x = s;    a.y = 0.0f;
    v2f b; b.x = 1.0f; b.y = 1.0f;
    v8f cacc = {};
    cacc = __builtin_amdgcn_wmma_f32_16x16x4_f32(
        /*neg_a=*/false, a, /*neg_b=*/false, b,
        /*c_mod=*/(short)0, cacc, /*reuse_a=*/false, /*reuse_b=*/false);

    float colsum = cacc[0] + cacc[1] + cacc[2] + cacc[3]
                 + cacc[4] + cacc[5] + cacc[6] + cacc[7];

    float low  = __shfl(colsum, 0,  32);  // rows 0..7  summed over all lanes
    float high = __shfl(colsum, 16, 32);  // rows 8..15 summed over all lanes

    if (lane == 0) out[0] = (low + high) * inv_batch;
}

extern "C" void kernel_launch(void* const* d_in, const int* in_sizes, int n_in,
                              void* d_out, int out_size, void* d_ws, size_t ws_size,
                              hipStream_t stream) {
    const float* preds   = (const float*)d_in[0];
    const float* targets = (const float*)d_in[1];
    float* partials = (float*)d_ws;          // P1_BLOCKS floats = 4 KB
    float* out      = (float*)d_out;

    yolo_pass1<<<P1_BLOCKS, P1_THREADS, 0, stream>>>(preds, targets, partials);
    yolo_reduce<<<1, 32, 0, stream>>>(partials, P1_BLOCKS, out, 1.0f / (float)NBATCH);
}